// HGCN_3212635537967
// MI455X (gfx1250) — compile-verified
//
#include <hip/hip_runtime.h>
#include <math.h>

typedef __attribute__((ext_vector_type(2))) float v2f;
typedef __attribute__((ext_vector_type(8))) float v8f;

// ---------------------------------------------------------------------------
// Generic kNN: one workgroup per (query, batch). Distances d = 2<q,x> - |q|^2 - |x|^2
// (same formula as reference), stable top-k (ties -> lowest index, like lax.top_k).
// ---------------------------------------------------------------------------
__global__ void knn_kernel(const float* __restrict__ X, long sb, long sn, long sc,
                           int C, int Npts, const int* __restrict__ qidx,
                           int k, int* __restrict__ out) {
    __shared__ float distL[2048];
    __shared__ float qv[64];
    int b = blockIdx.y, qi = blockIdx.x, tid = threadIdx.x;
    int q = qidx ? qidx[b * gridDim.x + qi] : qi;
    for (int c = tid; c < C; c += 256) qv[c] = X[b * sb + (long)q * sn + (long)c * sc];
    __syncthreads();
    float xxq = 0.f;
    for (int c = 0; c < C; ++c) xxq += qv[c] * qv[c];
    for (int j = tid; j < Npts; j += 256) {
        float inner = 0.f, xxj = 0.f;
        for (int c = 0; c < C; ++c) {
            float v = X[b * sb + (long)j * sn + (long)c * sc];
            inner += qv[c] * v; xxj += v * v;
        }
        distL[j] = 2.f * inner - xxq - xxj;
    }
    __syncthreads();
    if (tid == 0) {
        float kv[32]; int ki[32]; int cnt = 0;
        for (int j = 0; j < Npts; ++j) {
            float v = distL[j];
            if (cnt < k) {
                int p = cnt++;
                while (p > 0 && kv[p - 1] < v) { kv[p] = kv[p - 1]; ki[p] = ki[p - 1]; --p; }
                kv[p] = v; ki[p] = j;
            } else if (v > kv[k - 1]) {
                int p = k - 1;
                while (p > 0 && kv[p - 1] < v) { kv[p] = kv[p - 1]; ki[p] = ki[p - 1]; --p; }
                kv[p] = v; ki[p] = j;
            }
        }
        for (int i = 0; i < k; ++i) out[(b * gridDim.x + qi) * k + i] = ki[i];
    }
}

// ---------------------------------------------------------------------------
// Fused edge conv. Workgroup = (b, 4 points) -> 80 feature rows of width KF
// (KF = 2C padded to mult of 4). 8 waves run 5x4 WMMA f32 16x16 tile jobs.
// mode 0: write per-workgroup BN partial sums. mode 1: BN + LeakyReLU(0.2) + max over k.
// ---------------------------------------------------------------------------
template <int C, int KF>
__global__ void edgeconv_kernel(const float* __restrict__ X, long sb, long sn, long sc,
                                const int* __restrict__ idx, const float* __restrict__ W,
                                int N, int mode, float* __restrict__ partials,
                                const float* __restrict__ st, float* __restrict__ Xout) {
    __shared__ float featL[80 * KF];
    __shared__ float yL[80 * 64];
    const int KC = 2 * C;
    int b = blockIdx.y, nbase = blockIdx.x * 4, tid = threadIdx.x;
    for (int e = tid; e < 80 * KF; e += 256) {
        int r = e / KF, c = e % KF;
        int n = nbase + r / 20, kk = r % 20;
        int j = idx[(b * N + n) * 20 + kk];
        float v = 0.f;
        if (c < C)       v = X[b * sb + (long)j * sn + (long)c * sc] -
                             X[b * sb + (long)n * sn + (long)c * sc];
        else if (c < KC) v = X[b * sb + (long)n * sn + (long)(c - C) * sc];
        featL[r * KF + c] = v;
    }
    __syncthreads();
    int wave = tid >> 5, lane = tid & 31;
    int koff = (lane >> 4) * 2;
    for (int job = wave; job < 20; job += 8) {
        int rt = job >> 2, ct = job & 3;
        v8f acc = {};
        int rowl = rt * 16 + (lane & 15);
        int coll = ct * 16 + (lane & 15);
        for (int k0 = 0; k0 < KF; k0 += 4) {
            int ka = k0 + koff;
            // LDS 64-bit load: (rowl*KF + ka) is even -> 8B aligned
            v2f a = *(const v2f*)(featL + rowl * KF + ka);
            // Weight pair load; guard folds away when KF == 2C (conv2).
            bool okw = (KF == KC) || (ka < KC);
            const float* wp = W + (long)coll * KC + (okw ? ka : 0);
            v2f bv = *(const v2f*)wp;
            bv.x = okw ? bv.x : 0.f;
            bv.y = okw ? bv.y : 0.f;
            acc = __builtin_amdgcn_wmma_f32_16x16x4_f32(false, a, false, bv,
                                                        (short)0, acc, false, false);
        }
        for (int v = 0; v < 8; ++v) {
            int rr = rt * 16 + v + ((lane >> 4) << 3);
            yL[rr * 64 + coll] = acc[v];
        }
    }
    __syncthreads();
    if (mode == 0) {
        if (tid < 64) {
            float s = 0.f, s2 = 0.f;
            for (int r = 0; r < 80; ++r) { float v = yL[r * 64 + tid]; s += v; s2 += v * v; }
            int wg = blockIdx.y * gridDim.x + blockIdx.x;
            partials[wg * 128 + tid] = s;
            partials[wg * 128 + 64 + tid] = s2;
        }
    } else {
        int nl = tid >> 6, ch = tid & 63;
        float scale = st[ch * 2], shift = st[ch * 2 + 1];
        float m = -3.4e38f;
        for (int kk = 0; kk < 20; ++kk) {
            float v = yL[(nl * 20 + kk) * 64 + ch] * scale + shift;
            v = v > 0.f ? v : 0.2f * v;
            m = fmaxf(m, v);
        }
        Xout[((long)(b * N + nbase + nl)) * 64 + ch] = m;
    }
}

// Deterministic reduction of edge-conv BN partials -> scale/shift (64 channels).
__global__ void reduce_conv_stats(const float* __restrict__ partials, int nwg, long M,
                                  const float* __restrict__ gamma,
                                  const float* __restrict__ beta, float* __restrict__ st) {
    int ch = threadIdx.x;  // 64 threads
    double s = 0.0, s2 = 0.0;
    for (int w = 0; w < nwg; ++w) {
        s += (double)partials[w * 128 + ch];
        s2 += (double)partials[w * 128 + 64 + ch];
    }
    double mean = s / (double)M;
    double var = s2 / (double)M - mean * mean;
    float scale = gamma[ch] * (float)(1.0 / sqrt(var + 1e-5));
    st[ch * 2] = scale;
    st[ch * 2 + 1] = beta[ch] - (float)mean * scale;
}

// ---------------------------------------------------------------------------
// Generic WMMA fp32 GEMM: Y[R,O] = A[R,K] @ W[O,K]^T (+bias). One wave per 16x16 tile.
// Out-of-range rows/cols: load from clamped (valid) address, zero via cndmask.
// ---------------------------------------------------------------------------
__global__ void gemm_wmma(const float* __restrict__ A, const float* __restrict__ W,
                          const float* __restrict__ bias, float* __restrict__ Y,
                          int R, int K, int O) {
    int rbase = blockIdx.x * 16, cbase = blockIdx.y * 16;
    int lane = threadIdx.x;
    int row = rbase + (lane & 15);
    int col = cbase + (lane & 15);
    int koff = (lane >> 4) * 2;
    bool rok = row < R, cok = col < O;
    int rs = rok ? row : 0;     // clamped, always-valid addresses
    int cs = cok ? col : 0;
    const float* Ap = A + (long)rs * K + koff;
    const float* Wp = W + (long)cs * K + koff;
    __builtin_prefetch(Wp, 0, 0);
    v8f acc = {};
    for (int k0 = 0; k0 < K; k0 += 4) {
        v2f a = *(const v2f*)(Ap + k0);   // global_load_b64 (koff even, K mult of 4)
        v2f bv = *(const v2f*)(Wp + k0);
        a.x = rok ? a.x : 0.f;
        a.y = rok ? a.y : 0.f;
        bv.x = cok ? bv.x : 0.f;
        bv.y = cok ? bv.y : 0.f;
        acc = __builtin_amdgcn_wmma_f32_16x16x4_f32(false, a, false, bv,
                                                    (short)0, acc, false, false);
    }
    float bb = (cok && bias) ? bias[col] : 0.f;
    for (int v = 0; v < 8; ++v) {
        int r = rbase + v + ((lane >> 4) << 3);
        int c = cbase + (lane & 15);
        if (r < R && c < O) Y[(long)r * O + c] = acc[v] + bb;
    }
}

// Per-group BN stats over (B,S): rows laid out ((b*G+g)*S+s). One block per g.
__global__ void stats_kernel(const float* __restrict__ Y, int B, int G, int S, int O,
                             const float* __restrict__ gamma, const float* __restrict__ beta,
                             float* __restrict__ st) {
    int g = blockIdx.x;
    for (int o = threadIdx.x; o < O; o += blockDim.x) {
        double s = 0.0, s2 = 0.0;
        for (int b = 0; b < B; ++b)
            for (int ss = 0; ss < S; ++ss) {
                float v = Y[((long)((b * G + g) * S + ss)) * O + o];
                s += (double)v; s2 += (double)v * (double)v;
            }
        double M = (double)(B * S);
        double mean = s / M;
        double var = s2 / M - mean * mean;
        float scale = gamma[o] * (float)(1.0 / sqrt(var + 1e-5));
        st[(g * O + o) * 2] = scale;
        st[(g * O + o) * 2 + 1] = beta[o] - (float)mean * scale;
    }
}

__global__ void bnact_kernel(float* __restrict__ Y, long total, int G, int S, int O,
                             const float* __restrict__ st, int relu) {
    long t = (long)blockIdx.x * blockDim.x + threadIdx.x;
    if (t >= total) return;
    long row = t / O; int o = (int)(t % O);
    int g = (int)((row / S) % G);
    float v = Y[t] * st[(g * O + o) * 2] + st[(g * O + o) * 2 + 1];
    if (relu) v = fmaxf(v, 0.f);
    Y[t] = v;
}

__global__ void maxs_kernel(const float* __restrict__ Y, int B, int G, int S, int O,
                            float* __restrict__ out) {
    int total = B * G * O;
    int t = blockIdx.x * blockDim.x + threadIdx.x;
    if (t >= total) return;
    int o = t % O; int bg = t / O;
    float m = -3.4e38f;
    for (int s = 0; s < S; ++s) m = fmaxf(m, Y[((long)(bg * S + s)) * O + o]);
    out[t] = m;
}

// Farthest point sampling (start index 0, argmax ties -> lowest index).
__global__ void fps_kernel(const float* __restrict__ X, long sb, long sn, long sc,
                           int Npts, int S, int* __restrict__ out) {
    __shared__ float distL[2048];
    __shared__ float cvec[3];
    __shared__ float rv[256];
    __shared__ int ri[256];
    int b = blockIdx.x, tid = threadIdx.x;
    for (int j = tid; j < Npts; j += 256) distL[j] = 1e10f;
    if (tid == 0) out[b * S] = 0;
    int far = 0;
    __syncthreads();
    for (int s = 1; s < S; ++s) {
        if (tid < 3) cvec[tid] = X[b * sb + (long)far * sn + (long)tid * sc];
        __syncthreads();
        for (int j = tid; j < Npts; j += 256) {
            float d = 0.f;
            for (int c = 0; c < 3; ++c) {
                float df = X[b * sb + (long)j * sn + (long)c * sc] - cvec[c];
                d += df * df;
            }
            distL[j] = fminf(distL[j], d);
        }
        __syncthreads();
        float bv = -3.4e38f; int bi = 0x7fffffff;
        for (int j = tid; j < Npts; j += 256) {
            float v = distL[j];
            if (v > bv) { bv = v; bi = j; }
        }
        rv[tid] = bv; ri[tid] = bi;
        __syncthreads();
        for (int off = 128; off > 0; off >>= 1) {
            if (tid < off) {
                if (rv[tid + off] > rv[tid] ||
                    (rv[tid + off] == rv[tid] && ri[tid + off] < ri[tid])) {
                    rv[tid] = rv[tid + off]; ri[tid] = ri[tid + off];
                }
            }
            __syncthreads();
        }
        far = ri[0];
        if (tid == 0) out[b * S + s] = far;
        __syncthreads();
    }
}

// pc[b,s,c] = X[b, fidx[b,s], c]  (row-major (B,S,3) output)
__global__ void gather_coords(const float* __restrict__ X, long sb, long sn, long sc,
                              const int* __restrict__ fidx, int S, float* __restrict__ out,
                              int total) {
    int t = blockIdx.x * blockDim.x + threadIdx.x;
    if (t >= total) return;
    int c = t % 3; int r = t / 3; int g = r % S; int b = r / S;
    out[t] = X[b * sb + (long)fidx[b * S + g] * sn + (long)c * sc];
}

// out[b,g,s,c] = F[b, idx[b,g,s], c]   (F row-major (B,Nf,C))
__global__ void gather_feat(const float* __restrict__ F, int Nf, int C,
                            const int* __restrict__ idxb, int G, int S,
                            float* __restrict__ out, long total) {
    long t = (long)blockIdx.x * blockDim.x + threadIdx.x;
    if (t >= total) return;
    int c = (int)(t % C); long r = t / C;
    int s = (int)(r % S); long r2 = r / S;
    int g = (int)(r2 % G); int b = (int)(r2 / G);
    int j = idxb[(b * G + g) * S + s];
    out[t] = F[((long)(b * Nf + j)) * C + c];
}

// ---------------------------------------------------------------------------
extern "C" void kernel_launch(void* const* d_in, const int* in_sizes, int n_in,
                              void* d_out, int out_size, void* d_ws, size_t ws_size,
                              hipStream_t stream) {
    (void)in_sizes; (void)n_in; (void)out_size; (void)ws_size;
    const int B = 8, N = 2048, KK = 20;

    const float* x       = (const float*)d_in[0];
    const float* conv1_w = (const float*)d_in[1];
    const float* conv1_g = (const float*)d_in[2];
    const float* conv1_b = (const float*)d_in[3];
    const float* conv2_w = (const float*)d_in[4];
    const float* conv2_g = (const float*)d_in[5];
    const float* conv2_b = (const float*)d_in[6];
    const float* h1_w1 = (const float*)d_in[7];
    const float* h1_g1 = (const float*)d_in[8];
    const float* h1_b1 = (const float*)d_in[9];
    const float* h1_w2 = (const float*)d_in[10];
    const float* h1_g2 = (const float*)d_in[11];
    const float* h1_b2 = (const float*)d_in[12];
    const float* h2_w1 = (const float*)d_in[13];
    const float* h2_g1 = (const float*)d_in[14];
    const float* h2_b1 = (const float*)d_in[15];
    const float* h2_w2 = (const float*)d_in[16];
    const float* h2_g2 = (const float*)d_in[17];
    const float* h2_b2 = (const float*)d_in[18];
    const float* gl_w1 = (const float*)d_in[19];
    const float* gl_g1 = (const float*)d_in[20];
    const float* gl_b1 = (const float*)d_in[21];
    const float* gl_w2 = (const float*)d_in[22];
    const float* gl_g2 = (const float*)d_in[23];
    const float* gl_b2 = (const float*)d_in[24];
    const float* cls_w1 = (const float*)d_in[25];
    const float* cls_g1 = (const float*)d_in[26];
    const float* cls_b1 = (const float*)d_in[27];
    const float* cls_w2 = (const float*)d_in[28];
    const float* cls_bias2 = (const float*)d_in[29];
    const float* cls_g2 = (const float*)d_in[30];
    const float* cls_b2 = (const float*)d_in[31];
    const float* cls_w3 = (const float*)d_in[32];
    const float* cls_bias3 = (const float*)d_in[33];

    char* base = (char*)d_ws;
    size_t off = 0;
    auto alloc = [&](size_t bytes) -> void* {
        off = (off + 255) & ~(size_t)255;
        void* p = base + off;
        off += bytes;
        return p;
    };
    int* idx1 = (int*)alloc((size_t)B * N * KK * 4);
    int* idx2 = (int*)alloc((size_t)B * N * KK * 4);
    int* idx3 = (int*)alloc((size_t)B * 32 * 32 * 4);
    int* idx4 = (int*)alloc((size_t)B * 8 * 8 * 4);
    int* fidx1 = (int*)alloc((size_t)B * 32 * 4);
    int* fidx2 = (int*)alloc((size_t)B * 8 * 4);
    float* x1 = (float*)alloc((size_t)B * N * 64 * 4);
    float* x2 = (float*)alloc((size_t)B * N * 64 * 4);
    float* pc1T = (float*)alloc((size_t)B * 32 * 3 * 4);
    float* partials = (float*)alloc((size_t)(N / 4) * B * 128 * 4);
    float* stbuf = (float*)alloc((size_t)32 * 128 * 2 * 4);
    float* YA = (float*)alloc((size_t)8192 * 128 * 4);
    float* YB = (float*)alloc((size_t)8192 * 128 * 4);
    float* h1 = (float*)alloc((size_t)B * 32 * 32 * 64 * 4);
    float* f1 = (float*)alloc((size_t)B * 32 * 128 * 4);
    float* h2 = (float*)alloc((size_t)B * 8 * 8 * 128 * 4);
    float* f2 = (float*)alloc((size_t)B * 8 * 256 * 4);
    float* gb = (float*)alloc((size_t)B * 512 * 4);
    float* c1 = (float*)alloc((size_t)B * 512 * 4);
    float* c2 = (float*)alloc((size_t)B * 256 * 4);

    const long xsb = 3L * N, xsn = 1L, xsc = (long)N;   // x is (B,3,N)
    const int nwg = (N / 4) * B;

    // ---- edge conv 1 ----
    knn_kernel<<<dim3(N, B), 256, 0, stream>>>(x, xsb, xsn, xsc, 3, N, nullptr, KK, idx1);
    edgeconv_kernel<3, 8><<<dim3(N / 4, B), 256, 0, stream>>>(
        x, xsb, xsn, xsc, idx1, conv1_w, N, 0, partials, nullptr, nullptr);
    reduce_conv_stats<<<1, 64, 0, stream>>>(partials, nwg, (long)B * N * KK, conv1_g, conv1_b, stbuf);
    edgeconv_kernel<3, 8><<<dim3(N / 4, B), 256, 0, stream>>>(
        x, xsb, xsn, xsc, idx1, conv1_w, N, 1, nullptr, stbuf, x1);

    // ---- edge conv 2 ----
    knn_kernel<<<dim3(N, B), 256, 0, stream>>>(x1, (long)N * 64, 64L, 1L, 64, N, nullptr, KK, idx2);
    edgeconv_kernel<64, 128><<<dim3(N / 4, B), 256, 0, stream>>>(
        x1, (long)N * 64, 64L, 1L, idx2, conv2_w, N, 0, partials, nullptr, nullptr);
    reduce_conv_stats<<<1, 64, 0, stream>>>(partials, nwg, (long)B * N * KK, conv2_g, conv2_b, stbuf);
    edgeconv_kernel<64, 128><<<dim3(N / 4, B), 256, 0, stream>>>(
        x1, (long)N * 64, 64L, 1L, idx2, conv2_w, N, 1, nullptr, stbuf, x2);

    // ---- hierarchy level 1: FPS 32, group 32 ----
    fps_kernel<<<B, 256, 0, stream>>>(x, xsb, xsn, xsc, N, 32, fidx1);
    gather_coords<<<(B * 32 * 3 + 255) / 256, 256, 0, stream>>>(x, xsb, xsn, xsc, fidx1, 32,
                                                                pc1T, B * 32 * 3);
    knn_kernel<<<dim3(32, B), 256, 0, stream>>>(x, xsb, xsn, xsc, 3, N, fidx1, 32, idx3);
    gather_feat<<<(B * 32 * 32 * 64 + 255) / 256, 256, 0, stream>>>(
        x2, N, 64, idx3, 32, 32, h1, (long)B * 32 * 32 * 64);

    // local embedder 1: (8192 x 64) -> 128 -> 128, BN per group, max over S=32
    gemm_wmma<<<dim3(512, 8), 32, 0, stream>>>(h1, h1_w1, nullptr, YA, 8192, 64, 128);
    stats_kernel<<<32, 256, 0, stream>>>(YA, B, 32, 32, 128, h1_g1, h1_b1, stbuf);
    bnact_kernel<<<(int)(((long)8192 * 128 + 255) / 256), 256, 0, stream>>>(
        YA, (long)8192 * 128, 32, 32, 128, stbuf, 1);
    gemm_wmma<<<dim3(512, 8), 32, 0, stream>>>(YA, h1_w2, nullptr, YB, 8192, 128, 128);
    stats_kernel<<<32, 256, 0, stream>>>(YB, B, 32, 32, 128, h1_g2, h1_b2, stbuf);
    bnact_kernel<<<(int)(((long)8192 * 128 + 255) / 256), 256, 0, stream>>>(
        YB, (long)8192 * 128, 32, 32, 128, stbuf, 1);
    maxs_kernel<<<(B * 32 * 128 + 255) / 256, 256, 0, stream>>>(YB, B, 32, 32, 128, f1);

    // ---- hierarchy level 2: FPS 8, group 8 (on pc1) ----
    fps_kernel<<<B, 256, 0, stream>>>(pc1T, 96L, 3L, 1L, 32, 8, fidx2);
    knn_kernel<<<dim3(8, B), 256, 0, stream>>>(pc1T, 96L, 3L, 1L, 3, 32, fidx2, 8, idx4);
    gather_feat<<<(B * 8 * 8 * 128 + 255) / 256, 256, 0, stream>>>(
        f1, 32, 128, idx4, 8, 8, h2, (long)B * 8 * 8 * 128);

    // local embedder 2: (512 x 128) -> 256 -> 256, BN per group, max over S=8
    gemm_wmma<<<dim3(32, 16), 32, 0, stream>>>(h2, h2_w1, nullptr, YA, 512, 128, 256);
    stats_kernel<<<8, 256, 0, stream>>>(YA, B, 8, 8, 256, h2_g1, h2_b1, stbuf);
    bnact_kernel<<<(512 * 256 + 255) / 256, 256, 0, stream>>>(YA, 512L * 256, 8, 8, 256, stbuf, 1);
    gemm_wmma<<<dim3(32, 16), 32, 0, stream>>>(YA, h2_w2, nullptr, YB, 512, 256, 256);
    stats_kernel<<<8, 256, 0, stream>>>(YB, B, 8, 8, 256, h2_g2, h2_b2, stbuf);
    bnact_kernel<<<(512 * 256 + 255) / 256, 256, 0, stream>>>(YB, 512L * 256, 8, 8, 256, stbuf, 1);
    maxs_kernel<<<(B * 8 * 256 + 255) / 256, 256, 0, stream>>>(YB, B, 8, 8, 256, f2);

    // ---- global embedder: (64 x 256) -> 512 -> 512, BN over (B,S), max over S=8 ----
    gemm_wmma<<<dim3(4, 32), 32, 0, stream>>>(f2, gl_w1, nullptr, YA, 64, 256, 512);
    stats_kernel<<<1, 256, 0, stream>>>(YA, B, 1, 8, 512, gl_g1, gl_b1, stbuf);
    bnact_kernel<<<(64 * 512 + 255) / 256, 256, 0, stream>>>(YA, 64L * 512, 1, 8, 512, stbuf, 1);
    gemm_wmma<<<dim3(4, 32), 32, 0, stream>>>(YA, gl_w2, nullptr, YB, 64, 512, 512);
    stats_kernel<<<1, 256, 0, stream>>>(YB, B, 1, 8, 512, gl_g2, gl_b2, stbuf);
    bnact_kernel<<<(64 * 512 + 255) / 256, 256, 0, stream>>>(YB, 64L * 512, 1, 8, 512, stbuf, 1);
    maxs_kernel<<<(B * 512 + 255) / 256, 256, 0, stream>>>(YB, B, 1, 8, 512, gb);

    // ---- classifier head ----
    gemm_wmma<<<dim3(1, 32), 32, 0, stream>>>(gb, cls_w1, nullptr, c1, 8, 512, 512);
    stats_kernel<<<1, 256, 0, stream>>>(c1, B, 1, 1, 512, cls_g1, cls_b1, stbuf);
    bnact_kernel<<<(8 * 512 + 255) / 256, 256, 0, stream>>>(c1, 8L * 512, 1, 1, 512, stbuf, 0);
    gemm_wmma<<<dim3(1, 16), 32, 0, stream>>>(c1, cls_w2, cls_bias2, c2, 8, 512, 256);
    stats_kernel<<<1, 256, 0, stream>>>(c2, B, 1, 1, 256, cls_g2, cls_b2, stbuf);
    bnact_kernel<<<(8 * 256 + 255) / 256, 256, 0, stream>>>(c2, 8L * 256, 1, 1, 256, stbuf, 0);
    gemm_wmma<<<dim3(1, 3), 32, 0, stream>>>(c2, cls_w3, cls_bias3, (float*)d_out, 8, 256, 40);
}